// GumbelVectorQuantizer_76416058131072
// MI455X (gfx1250) — compile-verified
//
#include <hip/hip_runtime.h>

typedef __attribute__((ext_vector_type(2))) float v2f;
typedef __attribute__((ext_vector_type(8))) float v8f;

static constexpr int kD = 512;   // input dim
static constexpr int kV = 8192;  // codebook size
static constexpr int kM = 8192;  // B*T = 8*1024

// ---------------------------------------------------------------------------
// K1: logits[M,V] = X[M,D] @ W[V,D]^T + bias  via V_WMMA_F32_16X16X4_F32
// grid = (V/64, M/256), block = 256 threads (8 waves).
// Each wave computes a 32(M) x 64(N) tile: 2 A frags, 4 B frags, 8 C frags,
// 8 WMMAs per K-step, 128 K-steps (K advances 4 per WMMA).
//
// fp32 WMMA fragment layout (ISA 7.12.2):
//   A 16x4 : lane holds row = lane&15; VGPR0/1 = k = 2*(lane>>4) + {0,1}
//   B 4x16 : lane holds col = lane&15; VGPR0/1 = k = 2*(lane>>4) + {0,1}
//   C 16x16: element r of v8f = (M = r + 8*(lane>>4), N = lane&15)
// ---------------------------------------------------------------------------
__global__ __launch_bounds__(256) void k_gemm_logits(
    const float* __restrict__ X, const float* __restrict__ W,
    const float* __restrict__ bias, float* __restrict__ logits) {
  const int lane = threadIdx.x & 31;
  const int wave = threadIdx.x >> 5;
  const int r    = lane & 15;
  const int h    = lane >> 4;  // which k-half (0 or 1)

  const int nBase = blockIdx.x * 64;
  const int mBase = blockIdx.y * 256 + wave * 32;

  const float* ap0 = X + (size_t)(mBase + r) * kD + 2 * h;
  const float* ap1 = ap0 + (size_t)16 * kD;
  const float* bp0 = W + (size_t)(nBase + r) * kD + 2 * h;
  const float* bp1 = bp0 + (size_t)16 * kD;
  const float* bp2 = bp0 + (size_t)32 * kD;
  const float* bp3 = bp0 + (size_t)48 * kD;

  v8f c00 = {}, c01 = {}, c02 = {}, c03 = {};
  v8f c10 = {}, c11 = {}, c12 = {}, c13 = {};

#pragma unroll 4
  for (int k = 0; k < kD; k += 4) {
    v2f a0 = *(const v2f*)(ap0 + k);
    v2f a1 = *(const v2f*)(ap1 + k);
    v2f b0 = *(const v2f*)(bp0 + k);
    v2f b1 = *(const v2f*)(bp1 + k);
    v2f b2 = *(const v2f*)(bp2 + k);
    v2f b3 = *(const v2f*)(bp3 + k);
    c00 = __builtin_amdgcn_wmma_f32_16x16x4_f32(false, a0, false, b0, (short)0, c00, false, false);
    c01 = __builtin_amdgcn_wmma_f32_16x16x4_f32(false, a0, false, b1, (short)0, c01, false, false);
    c02 = __builtin_amdgcn_wmma_f32_16x16x4_f32(false, a0, false, b2, (short)0, c02, false, false);
    c03 = __builtin_amdgcn_wmma_f32_16x16x4_f32(false, a0, false, b3, (short)0, c03, false, false);
    c10 = __builtin_amdgcn_wmma_f32_16x16x4_f32(false, a1, false, b0, (short)0, c10, false, false);
    c11 = __builtin_amdgcn_wmma_f32_16x16x4_f32(false, a1, false, b1, (short)0, c11, false, false);
    c12 = __builtin_amdgcn_wmma_f32_16x16x4_f32(false, a1, false, b2, (short)0, c12, false, false);
    c13 = __builtin_amdgcn_wmma_f32_16x16x4_f32(false, a1, false, b3, (short)0, c13, false, false);
  }

  float bsc0 = bias[nBase + r];
  float bsc1 = bias[nBase + 16 + r];
  float bsc2 = bias[nBase + 32 + r];
  float bsc3 = bias[nBase + 48 + r];

#define STORE_FRAG(CC, I, J, BJ)                                             \
  {                                                                          \
    size_t rb = (size_t)(mBase + (I)*16 + 8 * h) * kV + nBase + 16 * (J) + r;\
    _Pragma("unroll") for (int rr = 0; rr < 8; ++rr)                         \
        logits[rb + (size_t)rr * kV] = (CC)[rr] + (BJ);                      \
  }
  STORE_FRAG(c00, 0, 0, bsc0)
  STORE_FRAG(c01, 0, 1, bsc1)
  STORE_FRAG(c02, 0, 2, bsc2)
  STORE_FRAG(c03, 0, 3, bsc3)
  STORE_FRAG(c10, 1, 0, bsc0)
  STORE_FRAG(c11, 1, 1, bsc1)
  STORE_FRAG(c12, 1, 2, bsc2)
  STORE_FRAG(c13, 1, 3, bsc3)
#undef STORE_FRAG
}

// ---------------------------------------------------------------------------
// K2: per-row softmax + argmax. One 256-thread block per row (8192 rows).
// Row slab (32KB) stays L2-hot between the three sweeps.
// ---------------------------------------------------------------------------
__global__ __launch_bounds__(256) void k_softmax_argmax(
    const float* __restrict__ logits, float* __restrict__ soft,
    float* __restrict__ idx_f, int* __restrict__ idx_i) {
  const int row = blockIdx.x;
  const int t = threadIdx.x;
  const float* lrow = logits + (size_t)row * kV;

  __shared__ float smax[256];
  __shared__ int sarg[256];
  __shared__ float ssum[256];

  float lmax = -3.402823466e+38f;
  int larg = 0;
  for (int v = t; v < kV; v += 256) {
    float x = lrow[v];
    if (x > lmax) { lmax = x; larg = v; }
  }
  smax[t] = lmax;
  sarg[t] = larg;
  __syncthreads();
  for (int s = 128; s > 0; s >>= 1) {
    if (t < s) {
      float o = smax[t + s];
      int oa = sarg[t + s];
      if (o > smax[t] || (o == smax[t] && oa < sarg[t])) {
        smax[t] = o;
        sarg[t] = oa;
      }
    }
    __syncthreads();
  }
  const float rmax = smax[0];

  float lsum = 0.f;
  for (int v = t; v < kV; v += 256) lsum += __expf(lrow[v] - rmax);
  ssum[t] = lsum;
  __syncthreads();
  for (int s = 128; s > 0; s >>= 1) {
    if (t < s) ssum[t] += ssum[t + s];
    __syncthreads();
  }
  const float inv = 1.0f / ssum[0];

  float* srow = soft + (size_t)row * kV;
  for (int v = t; v < kV; v += 256) srow[v] = __expf(lrow[v] - rmax) * inv;

  if (t == 0) {
    idx_f[row] = (float)sarg[0];
    idx_i[row] = sarg[0];
  }
}

// ---------------------------------------------------------------------------
// K3: quantized = codebook[idx]; per-block partial sum of (q - x)^2.
// grid = M*D/256 = 16384 blocks.
// ---------------------------------------------------------------------------
__global__ __launch_bounds__(256) void k_quant_loss(
    const float* __restrict__ X, const float* __restrict__ codebook,
    const int* __restrict__ idx, float* __restrict__ q,
    float* __restrict__ partial) {
  const int tid = threadIdx.x;
  const int i = blockIdx.x * 256 + tid;  // < M*D
  const int row = i >> 9;   // / kD
  const int d = i & (kD - 1);
  float cv = codebook[(size_t)idx[row] * kD + d];
  q[i] = cv;
  float diff = cv - X[i];

  __shared__ float s[256];
  s[tid] = diff * diff;
  __syncthreads();
  for (int st = 128; st > 0; st >>= 1) {
    if (tid < st) s[tid] += s[tid + st];
    __syncthreads();
  }
  if (tid == 0) partial[blockIdx.x] = s[0];
}

// ---------------------------------------------------------------------------
// K4: final deterministic reduction of 16384 partials -> mean.
// ---------------------------------------------------------------------------
__global__ __launch_bounds__(256) void k_loss_reduce(
    const float* __restrict__ partial, float* __restrict__ loss) {
  const int tid = threadIdx.x;
  float acc = 0.f;
  for (int i = tid; i < 16384; i += 256) acc += partial[i];
  __shared__ float s[256];
  s[tid] = acc;
  __syncthreads();
  for (int st = 128; st > 0; st >>= 1) {
    if (tid < st) s[tid] += s[tid + st];
    __syncthreads();
  }
  if (tid == 0) loss[0] = s[0] / (float)((size_t)kM * kD);
}

// ---------------------------------------------------------------------------
extern "C" void kernel_launch(void* const* d_in, const int* in_sizes, int n_in,
                              void* d_out, int out_size, void* d_ws,
                              size_t ws_size, hipStream_t stream) {
  (void)in_sizes; (void)n_in; (void)out_size; (void)ws_size;
  const float* X        = (const float*)d_in[0];  // [8,1024,512]
  const float* codebook = (const float*)d_in[1];  // [8192,512]
  const float* Wl       = (const float*)d_in[2];  // [8192,512]
  const float* bl       = (const float*)d_in[3];  // [8192]

  float* out   = (float*)d_out;
  float* q     = out;                      // [M,D]  4194304
  float* idx_f = q + (size_t)kM * kD;      // [M]    8192
  float* loss  = idx_f + kM;               // [1]
  float* soft  = loss + 1;                 // [M,V]  67108864
  float* logit = soft + (size_t)kM * kV;   // [M,V]  67108864

  int*   idx_i   = (int*)d_ws;             // [M] ints
  float* partial = (float*)d_ws + kM;      // [16384] floats

  k_gemm_logits<<<dim3(kV / 64, kM / 256), 256, 0, stream>>>(X, Wl, bl, logit);
  k_softmax_argmax<<<dim3(kM), 256, 0, stream>>>(logit, soft, idx_f, idx_i);
  k_quant_loss<<<dim3((kM * kD) / 256), 256, 0, stream>>>(X, codebook, idx_i, q,
                                                          partial);
  k_loss_reduce<<<1, 256, 0, stream>>>(partial, loss);
}